// LowpassDetector_47906065219752
// MI455X (gfx1250) — compile-verified
//
#include <hip/hip_runtime.h>
#include <cmath>
#include <complex>

// ---------------------------------------------------------------------------
// LowpassDetector: order-4 Butterworth IIR over [T, C] = [65536, 512] fp32.
// Block-parallel state-space decomposition:
//   pass 1: per-(block,channel) particular solution, 2 channels/lane (b64),
//           non-temporal x loads so y_p stays resident in the 192MB L2
//   pass 2: inter-block 4x4 state scan
//   pass 3: homogeneous correction  Y += H(16x4) x S0(4x16)  via
//           V_WMMA_F32_16X16X4_F32 (exact f32 shape match)
// ---------------------------------------------------------------------------

#define TT  65536   // timesteps
#define CC  512     // channels
#define BL  256     // block length (multiple of 16)
#define NB  (TT / BL)

typedef float v2f __attribute__((ext_vector_type(2)));
typedef float v8f __attribute__((ext_vector_type(8)));

struct Coef { float b[5]; float a[5]; };

// Workspace layout (floats):
//   H  : BL*4              impulse responses of the 4 state components
//   M  : 16                N^BL (4x4), column k = final state from e_k
//   Sp : NB*4*CC           per-block particular final states
//   S0 : NB*4*CC           per-block initial states after scan
#define WS_H   0
#define WS_M   (BL * 4)
#define WS_SP  (BL * 4 + 16)
#define WS_S0  (BL * 4 + 16 + NB * 4 * CC)

// --------------------------- init: build H and N^BL -------------------------
__global__ __launch_bounds__(32)
void lp_init(float* __restrict__ H, float* __restrict__ Mm, Coef cf) {
    int k = threadIdx.x;
    if (k < 4) {
        float z0 = (k == 0) ? 1.f : 0.f;
        float z1 = (k == 1) ? 1.f : 0.f;
        float z2 = (k == 2) ? 1.f : 0.f;
        float z3 = (k == 3) ? 1.f : 0.f;
        const float a1 = cf.a[1], a2 = cf.a[2], a3 = cf.a[3], a4 = cf.a[4];
        for (int n = 0; n < BL; ++n) {
            float y = z0;                 // homogeneous output response
            H[n * 4 + k] = y;
            float nz0 = z1 - a1 * y;
            float nz1 = z2 - a2 * y;
            float nz2 = z3 - a3 * y;
            float nz3 =    - a4 * y;
            z0 = nz0; z1 = nz1; z2 = nz2; z3 = nz3;
        }
        Mm[0 * 4 + k] = z0;
        Mm[1 * 4 + k] = z1;
        Mm[2 * 4 + k] = z2;
        Mm[3 * 4 + k] = z3;
    }
}

// ------------------ pass 1: block-local particular solution -----------------
// One lane = 2 adjacent channels (b64 traffic). x loads are non-temporal so
// the one-touch input stream does not evict y_p from L2 before pass 3.
__global__ __launch_bounds__(256)
void lp_particular(const v2f* __restrict__ x2, v2f* __restrict__ yp2,
                   v2f* __restrict__ Sp2, Coef cf) {
    const int t = threadIdx.x;                 // channel pair: 2t, 2t+1
    const int b = blockIdx.x;                  // time block
    const float b0 = cf.b[0], b1 = cf.b[1], b2 = cf.b[2], b3 = cf.b[3], b4 = cf.b[4];
    const float a1 = cf.a[1], a2 = cf.a[2], a3 = cf.a[3], a4 = cf.a[4];
    v2f z0 = {0.f, 0.f}, z1 = z0, z2 = z0, z3 = z0;
    const size_t base = (size_t)b * BL * (CC / 2) + t;
    #pragma unroll 4
    for (int n = 0; n < BL; ++n) {
        v2f xv = __builtin_nontemporal_load(x2 + base + (size_t)n * (CC / 2));
        v2f y  = b0 * xv + z0;
        v2f n0 = b1 * xv + z1 - a1 * y;
        v2f n1 = b2 * xv + z2 - a2 * y;
        v2f n2 = b3 * xv + z3 - a3 * y;
        v2f n3 = b4 * xv      - a4 * y;
        z0 = n0; z1 = n1; z2 = n2; z3 = n3;
        yp2[base + (size_t)n * (CC / 2)] = y;
    }
    Sp2[((size_t)b * 4 + 0) * (CC / 2) + t] = z0;
    Sp2[((size_t)b * 4 + 1) * (CC / 2) + t] = z1;
    Sp2[((size_t)b * 4 + 2) * (CC / 2) + t] = z2;
    Sp2[((size_t)b * 4 + 3) * (CC / 2) + t] = z3;
}

// ------------------- pass 2: sequential inter-block scan --------------------
__global__ __launch_bounds__(256)
void lp_scan(const float* __restrict__ Sp, const float* __restrict__ Mm,
             float* __restrict__ S0) {
    const int c = blockIdx.x * blockDim.x + threadIdx.x;
    float m[16];
    #pragma unroll
    for (int i = 0; i < 16; ++i) m[i] = Mm[i];
    float s0 = 0.f, s1 = 0.f, s2 = 0.f, s3 = 0.f;
    for (int b = 0; b < NB; ++b) {
        S0[((size_t)b * 4 + 0) * CC + c] = s0;
        S0[((size_t)b * 4 + 1) * CC + c] = s1;
        S0[((size_t)b * 4 + 2) * CC + c] = s2;
        S0[((size_t)b * 4 + 3) * CC + c] = s3;
        float p0 = Sp[((size_t)b * 4 + 0) * CC + c];
        float p1 = Sp[((size_t)b * 4 + 1) * CC + c];
        float p2 = Sp[((size_t)b * 4 + 2) * CC + c];
        float p3 = Sp[((size_t)b * 4 + 3) * CC + c];
        float t0 = m[0]  * s0 + m[1]  * s1 + m[2]  * s2 + m[3]  * s3 + p0;
        float t1 = m[4]  * s0 + m[5]  * s1 + m[6]  * s2 + m[7]  * s3 + p1;
        float t2 = m[8]  * s0 + m[9]  * s1 + m[10] * s2 + m[11] * s3 + p2;
        float t3 = m[12] * s0 + m[13] * s1 + m[14] * s2 + m[15] * s3 + p3;
        s0 = t0; s1 = t1; s2 = t2; s3 = t3;
    }
}

// -------- pass 3: WMMA correction  Y(16x16) += H(16x4) x S0(4x16) -----------
// One wave per 16x16 output tile. No divergence -> EXEC all ones at WMMA.
__global__ __launch_bounds__(256)
void lp_correct(float* __restrict__ y, const float* __restrict__ H,
                const float* __restrict__ S0) {
    const int lane = threadIdx.x & 31;
    const int wave = threadIdx.x >> 5;
    const int tile = blockIdx.x * 8 + wave;
    const int ctiles  = CC / 16;                 // 32
    const int rowTile = tile / ctiles;
    const int colTile = tile % ctiles;
    const int n0  = rowTile * 16;                // global timestep of tile row 0
    const int blk = n0 / BL;                     // time block (tiles never straddle)
    const int nl  = n0 % BL;                     // block-local timestep
    const int c0  = colTile * 16;

    const int mrow = lane & 15;                  // M (A) or N (B) index
    const int hi   = lane >> 4;                  // half-wave select
    const int kk   = hi * 2;                     // K pair held by this half-wave

    // A: 16x4 f32. VGPR0 holds K=kk, VGPR1 holds K=kk+1 (ISA 7.12.2).
    v2f A;
    A.x = H[(nl + mrow) * 4 + kk];
    A.y = H[(nl + mrow) * 4 + kk + 1];

    // B: 4x16 f32, row (K) striped across lanes within a VGPR.
    v2f B;
    B.x = S0[((size_t)blk * 4 + kk)     * CC + c0 + mrow];
    B.y = S0[((size_t)blk * 4 + kk + 1) * CC + c0 + mrow];

    // C: 16x16 f32 accumulator tile = particular solution y_p (L2-resident).
    v8f Cv;
    #pragma unroll
    for (int v = 0; v < 8; ++v)
        Cv[v] = y[((size_t)(n0 + v + 8 * hi)) * CC + c0 + mrow];

    v8f D = __builtin_amdgcn_wmma_f32_16x16x4_f32(
        /*neg_a=*/false, A, /*neg_b=*/false, B,
        /*c_mod=*/(short)0, Cv, /*reuse_a=*/false, /*reuse_b=*/false);

    #pragma unroll
    for (int v = 0; v < 8; ++v)
        y[((size_t)(n0 + v + 8 * hi)) * CC + c0 + mrow] = D[v];
}

// ----------------------------- host side ------------------------------------
static void butter_lowpass(Coef& cf) {
    const int order = 4;
    const double PI = 3.14159265358979323846;
    const double wn = 2.0 * 20e9 / 160e9;               // 0.25
    const double fs = 2.0;
    const double warped = 2.0 * fs * std::tan(PI * wn / fs);
    std::complex<double> p[4];
    for (int i = 0; i < order; ++i) {
        double mm = -order + 1 + 2 * i;                 // -3,-1,1,3
        p[i] = -std::exp(std::complex<double>(0.0, PI * mm / (2.0 * order))) * warped;
    }
    const double k   = std::pow(warped, order);
    const double fs2 = 2.0 * fs;
    std::complex<double> pz[4], prod(1.0, 0.0);
    for (int i = 0; i < order; ++i) {
        pz[i] = (fs2 + p[i]) / (fs2 - p[i]);
        prod *= (fs2 - p[i]);
    }
    const double kz = k * std::real(std::complex<double>(1.0, 0.0) / prod);
    const double bb[5] = {1.0, 4.0, 6.0, 4.0, 1.0};     // poly of 4 zeros at -1
    for (int i = 0; i < 5; ++i) cf.b[i] = (float)(kz * bb[i]);
    std::complex<double> ac[5] = {{1,0},{0,0},{0,0},{0,0},{0,0}};
    for (int i = 0; i < order; ++i)
        for (int j = i + 1; j >= 1; --j) ac[j] -= pz[i] * ac[j - 1];
    for (int i = 0; i < 5; ++i) cf.a[i] = (float)std::real(ac[i]);
}

extern "C" void kernel_launch(void* const* d_in, const int* in_sizes, int n_in,
                              void* d_out, int out_size, void* d_ws, size_t ws_size,
                              hipStream_t stream) {
    const float* x = (const float*)d_in[0];   // [T, C] fp32
    float* y       = (float*)d_out;           // [T, C] fp32
    float* ws      = (float*)d_ws;

    float* H  = ws + WS_H;
    float* Mm = ws + WS_M;
    float* Sp = ws + WS_SP;
    float* S0 = ws + WS_S0;

    Coef cf;
    butter_lowpass(cf);

    lp_init<<<1, 32, 0, stream>>>(H, Mm, cf);

    // 256 lanes x 2 channels = all 512 channels per block; NB time blocks.
    lp_particular<<<NB, 256, 0, stream>>>((const v2f*)x, (v2f*)y, (v2f*)Sp, cf);

    lp_scan<<<CC / 256, 256, 0, stream>>>(Sp, Mm, S0);

    const int tiles = (TT / 16) * (CC / 16);  // 131072 tiles, 8 waves/block
    lp_correct<<<tiles / 8, 256, 0, stream>>>(y, H, S0);
}